// LorentzianHardNegativeMiner_47450798686772
// MI455X (gfx1250) — compile-verified
//
#include <hip/hip_runtime.h>
#include <math.h>

typedef float v2f __attribute__((ext_vector_type(2)));
typedef float v8f __attribute__((ext_vector_type(8)));

#define B_DIM    1024
#define N_CAND   1024
#define D1       129     // 1 time + 128 spatial
#define KSEL     64
#define NTHREADS 256
#define DIST_EPS 1e-7f

__global__ __launch_bounds__(NTHREADS)
void lorentz_mine_kernel(const float* __restrict__ anchor,   // [B, 129]
                         const float* __restrict__ cand,     // [B, 1024, 129]
                         float* __restrict__ out_hn,         // [B, 64, 129]
                         float* __restrict__ out_dist)       // [B, 64]
{
    __shared__ float s_anchor[D1];
    __shared__ float s_dist[N_CAND];
    __shared__ int   s_idx[N_CAND];

    const int b   = blockIdx.x;
    const int tid = threadIdx.x;
    const float* __restrict__ anc = anchor + (size_t)b * D1;
    const float* __restrict__ cnd = cand   + (size_t)b * N_CAND * D1;

    for (int i = tid; i < D1; i += NTHREADS) s_anchor[i] = anc[i];
    __syncthreads();

    const int lane = tid & 31;
    const int wave = tid >> 5;
    const int h    = lane >> 4;    // half-wave (selects K pair per WMMA f32 layout)
    const int nl   = lane & 15;    // candidate within 16-wide tile

    const float a0 = s_anchor[0];

    // 8 waves x 8 tiles x 16 candidates = 1024 candidates per block.
    for (int t = 0; t < 8; ++t) {
        const int n = ((wave + (t << 3)) << 4) + nl;
        const float* __restrict__ crow = cnd + (size_t)n * D1;

        v8f acc = {};
        #pragma unroll 4
        for (int ks = 0; ks < 32; ++ks) {
            const int koff = (ks << 2) + (h << 1);   // k0 + 2h
            v2f a, bm;
            a.x  = s_anchor[1 + koff];
            a.y  = s_anchor[2 + koff];
            bm.x = crow[1 + koff];                   // one global_load_b64 per lane
            bm.y = crow[2 + koff];
            // A rows all replicate the anchor -> every C row = dot(anchor_sp, cand_sp)
            acc = __builtin_amdgcn_wmma_f32_16x16x4_f32(
                false, a, false, bm, (short)0, acc, false, false);
        }

        const float inner = acc[0] - a0 * crow[0];           // Lorentz inner product
        const float arg   = fmaxf(-inner, 1.0f + DIST_EPS);  // CURVATURE = 1
        const float dist  = acoshf(arg);
        if (h == 0) {            // lanes 0..15 hold the 16 unique results
            s_dist[n] = dist;
            s_idx[n]  = n;
        }
    }
    __syncthreads();

    // Bitonic sort ascending on (dist, idx): matches top_k order incl. tie-break.
    for (unsigned k = 2; k <= N_CAND; k <<= 1) {
        for (unsigned j = k >> 1; j > 0; j >>= 1) {
            #pragma unroll
            for (unsigned i0 = 0; i0 < N_CAND; i0 += NTHREADS) {
                const unsigned i   = i0 + tid;
                const unsigned ixj = i ^ j;
                if (ixj > i) {
                    const bool  up = ((i & k) == 0);
                    float di = s_dist[i], dj = s_dist[ixj];
                    int   ii = s_idx[i],  ij = s_idx[ixj];
                    const bool gt = (di > dj) || (di == dj && ii > ij);
                    if (gt == up) {
                        s_dist[i] = dj; s_dist[ixj] = di;
                        s_idx[i]  = ij; s_idx[ixj]  = ii;
                    }
                }
            }
            __syncthreads();
        }
    }

    // Emit top-64 distances.
    if (tid < KSEL) out_dist[(size_t)b * KSEL + tid] = s_dist[tid];

    // Gather the 64 hard-negative rows (129 f32 each).
    for (int t2 = tid; t2 < KSEL * D1; t2 += NTHREADS) {
        const int j = t2 / D1;
        const int d = t2 - j * D1;
        out_hn[((size_t)b * KSEL + j) * D1 + d] =
            cnd[(size_t)s_idx[j] * D1 + d];
    }
}

extern "C" void kernel_launch(void* const* d_in, const int* in_sizes, int n_in,
                              void* d_out, int out_size, void* d_ws, size_t ws_size,
                              hipStream_t stream) {
    (void)in_sizes; (void)n_in; (void)out_size; (void)d_ws; (void)ws_size;
    const float* anchor = (const float*)d_in[0];   // anchor_emb      [B, 129]
    const float* cand   = (const float*)d_in[1];   // candidate_negs  [B, 1024, 129]
    // d_in[2] is k (==64), fixed at compile time.

    float* out_hn   = (float*)d_out;                                   // [B, 64, 129]
    float* out_dist = (float*)d_out + (size_t)B_DIM * KSEL * D1;       // [B, 64]

    lorentz_mine_kernel<<<B_DIM, NTHREADS, 0, stream>>>(anchor, cand, out_hn, out_dist);
}